// OFDM_channel_9388798509340
// MI455X (gfx1250) — compile-verified
//
#include <hip/hip_runtime.h>
#include <math.h>
#include <stdint.h>

// ---------------------------------------------------------------------------
// OFDM channel pipeline for MI455X (gfx1250, wave32).
// FFT/IFFT-256 implemented as DFT matmuls on V_WMMA_F32_16X16X4_F32.
// LDS staging uses the CDNA5 async memory->LDS path (ASYNCcnt).
// ---------------------------------------------------------------------------

typedef float v2f __attribute__((ext_vector_type(2)));
typedef float v8f __attribute__((ext_vector_type(8)));

#define N_    256
#define P_    4
#define S_    16
#define M_    256
#define K_    16
#define L_    16
#define MK_   272            // M + K
#define SMK_  4624           // (S+1)*(M+K)
#define YLEN_ 4639           // SMK + L - 1
#define NP_   1024           // N*P

// d_out float offsets (return order: ip, isg, H, npw)
#define IP_OFF  0
#define IS_OFF  524288       // 256*4*1*256*2
#define H_OFF   8912896      // IS_OFF + 256*4*16*256*2
#define NPW_OFF 9437184      // H_OFF + 256*4*256*2

// workspace float offsets
#define TW_FLOATS   131072   // 16 tiles * 64 ksteps * 32 lanes * 4 floats
#define PILOT_OFF_F TW_FLOATS
#define FRAME_OFF_F (TW_FLOATS + 1024)

#define TWO_PI_256 0.0245436926061702596f   // 2*pi/256
#define PI_256     0.0122718463030851298f   // pi/256

__device__ __forceinline__ v8f wmma4(v2f a, v2f b, v8f c) {
  // D = A(16x4,f32) * B(4x16,f32) + C(16x16,f32)
  return __builtin_amdgcn_wmma_f32_16x16x4_f32(false, a, false, b,
                                               (short)0, c, false, false);
}

// Async global->LDS copy of one float2 (8 bytes), per-lane LDS address.
// Low 32 bits of a generic LDS pointer == wave-relative LDS byte offset
// (ISA 10.2: LDS aperture addresses map via addr[31:0]).
__device__ __forceinline__ void async_copy_f2(const float2* __restrict__ g,
                                              float2* lds) {
  uint32_t loff = (uint32_t)(uintptr_t)lds;
  uint64_t gaddr = (uint64_t)(uintptr_t)g;
  asm volatile("global_load_async_to_lds_b64 %0, %1, off"
               :: "v"(loff), "v"(gaddr)
               : "memory");
}
__device__ __forceinline__ void async_wait0() {
  asm volatile("s_wait_asynccnt 0x0" ::: "memory");
}

// --------------------------------------------------------------------------
// Twiddle table in exact WMMA A-fragment order.
// Entry e = ((mt*64 + ks)*32 + lane): float4{c(m,k0), c(m,k0+1), s(m,k0), s(m,k0+1)}
// m = mt*16 + (lane&15), k0 = ks*4 + ((lane>=16)?2:0), theta = 2*pi*m*k/256.
// --------------------------------------------------------------------------
__global__ void k_twiddle(float4* __restrict__ tw) {
  int e = blockIdx.x * 256 + threadIdx.x;      // < 32768
  int lane = e & 31;
  int ks = (e >> 5) & 63;
  int mt = e >> 11;
  int m = mt * 16 + (lane & 15);
  int k0 = ks * 4 + ((lane >> 4) << 1);
  float th0 = (float)((m * k0) & 255) * TWO_PI_256;
  float th1 = (float)((m * (k0 + 1)) & 255) * TWO_PI_256;
  float c0, s0, c1, s1;
  __sincosf(th0, &s0, &c0);
  __sincosf(th1, &s1, &c1);
  tw[e] = make_float4(c0, c1, s0, s1);
}

// --------------------------------------------------------------------------
// Pilot: ZadoffChu(1,256) (alpha==1), IFFT, then CP-extend to 272 samples.
// phase(k,m) = pi*(2*m*k - k*k)/256, period 512.
// --------------------------------------------------------------------------
__global__ void k_pilot(float2* __restrict__ pilot) {
  int m = threadIdx.x;                          // 0..255
  float sr = 0.f, si = 0.f;
  for (int k = 0; k < 256; ++k) {
    int r = (2 * m * k - k * k) % 512;
    if (r < 0) r += 512;
    float th = (float)r * PI_256;
    float s, c;
    __sincosf(th, &s, &c);
    sr += c; si += s;
  }
  sr *= (1.f / 256.f); si *= (1.f / 256.f);
  pilot[K_ + m] = make_float2(sr, si);
  if (m >= M_ - K_) pilot[m - (M_ - K_)] = make_float2(sr, si);
}

// --------------------------------------------------------------------------
// K1: per-symbol normalize + IFFT-256 via WMMA DFT + CP, write frame body.
// One block = 16 symbols = one (n,p). 8 waves x 2 freq-tiles each.
// Staging: async global->LDS with on-the-fly [sym][k] -> [k][sym] transpose.
// --------------------------------------------------------------------------
__global__ void k_ifft_frame(const float2* __restrict__ x,
                             const float4* __restrict__ tw,
                             float2* __restrict__ frame) {
  __shared__ float2 Xlds[256 * 17];   // [sample k][sym], padded stride 17
  __shared__ float red[256];
  __shared__ float scaleSh[16];
  const int tid = threadIdx.x;
  const int np = blockIdx.x;                         // (n*P+p)
  const size_t symBase = (size_t)np * 16;

  for (int sym = 0; sym < 16; ++sym) {
    async_copy_f2(&x[(symBase + sym) * 256 + tid], &Xlds[tid * 17 + sym]);
  }
  async_wait0();
  __syncthreads();

  {  // per-symbol power: scale = sqrt(0.5/power) = sqrt(256/sum|x|^2)
    int sym = tid >> 4, j = tid & 15;
    float p = 0.f;
    for (int i = 0; i < 16; ++i) {
      float2 v = Xlds[(j + i * 16) * 17 + sym];
      p += v.x * v.x + v.y * v.y;
    }
    red[sym * 16 + j] = p;
  }
  __syncthreads();
  if (tid < 16) {
    float s = 0.f;
    for (int j = 0; j < 16; ++j) s += red[tid * 16 + j];
    scaleSh[tid] = sqrtf(256.f / s);
  }
  __syncthreads();

  const int wv = tid >> 5, lane = tid & 31;
  const int symL = lane & 15;
  const int kHalf = (lane >> 4) << 1;
  const int rowAdd = (lane >= 16) ? 8 : 0;
  const float scl = scaleSh[symL] * (1.f / 256.f);   // ifft 1/M folded in
  float2* fw = frame + (size_t)np * SMK_ + (size_t)(1 + symL) * MK_;

  for (int tt = 0; tt < 2; ++tt) {
    const int mt = wv + tt * 8;
    v8f aCR = {0,0,0,0,0,0,0,0};
    v8f aSI = {0,0,0,0,0,0,0,0};
    v8f aCI = {0,0,0,0,0,0,0,0};
    v8f aSR = {0,0,0,0,0,0,0,0};
    const float4* twp = tw + (size_t)(mt * 64) * 32 + lane;
    for (int ks = 0; ks < 64; ++ks) {
      float4 w = twp[ks * 32];
      int kb = ks * 4 + kHalf;
      float2 b0 = Xlds[kb * 17 + symL];
      float2 b1 = Xlds[(kb + 1) * 17 + symL];
      v2f ac = {w.x, w.y}, as_ = {w.z, w.w};
      v2f br = {b0.x, b1.x}, bi = {b0.y, b1.y};
      aCR = wmma4(ac, br, aCR);
      aSI = wmma4(as_, bi, aSI);
      aCI = wmma4(ac, bi, aCI);
      aSR = wmma4(as_, br, aSR);
    }
    // ifft: (c + i s)/M  ->  Re = (cXr - sXi)/M, Im = (cXi + sXr)/M
#pragma unroll
    for (int r = 0; r < 8; ++r) {
      int m = mt * 16 + rowAdd + r;
      float reV = (aCR[r] - aSI[r]) * scl;
      float imV = (aCI[r] + aSR[r]) * scl;
      float2 v = make_float2(reV, imV);
      fw[K_ + m] = v;                               // body
      if (m >= M_ - K_) fw[m - (M_ - K_)] = v;      // cyclic prefix
    }
  }
}

// --------------------------------------------------------------------------
// K2: per-(n,p): clip -> 16-tap conv (registers) -> AWGN -> CFO rotate.
// Writes rotated time frame back to workspace and noise_pwr to d_out.
// --------------------------------------------------------------------------
__global__ void k_channel(const float2* __restrict__ cof,
                          const float2* __restrict__ noise,
                          const int* __restrict__ snrP,
                          const float2* __restrict__ pilot,
                          float2* __restrict__ frame,
                          float* __restrict__ out) {
  __shared__ float2 fr[SMK_];
  __shared__ float red[256];
  __shared__ float2 cofsh[L_];
  const int tid = threadIdx.x;
  const int np = blockIdx.x;

  if (tid < L_) cofsh[tid] = cof[np * L_ + tid];

  float psum = 0.f;
  for (int i = tid; i < SMK_; i += 256) {
    float2 v = (i < MK_) ? pilot[i] : frame[(size_t)np * SMK_ + i];
    fr[i] = v;
    psum += v.x * v.x + v.y * v.y;
  }
  red[tid] = psum;
  __syncthreads();
  for (int off = 128; off; off >>= 1) {
    if (tid < off) red[tid] += red[tid + off];
    __syncthreads();
  }
  float fpwr = red[0] / (2.f * (float)SMK_);
  float maxAmp = sqrtf(fpwr * 3.1622776601683794f);   // A = 10^0.5
  __syncthreads();

  for (int i = tid; i < SMK_; i += 256) {             // clip
    float2 v = fr[i];
    float amp = sqrtf(v.x * v.x + v.y * v.y);
    float sc = fminf(maxAmp / amp, 1.f);
    fr[i] = make_float2(v.x * sc, v.y * sc);
  }
  __syncthreads();

  float yr[19], yi[19];
  float ysum = 0.f;
#pragma unroll
  for (int it = 0; it < 19; ++it) {
    int t = tid + it * 256;
    float ar = 0.f, ai = 0.f;
    if (t < YLEN_) {
      int l0 = t - SMK_ + 1; if (l0 < 0) l0 = 0;
      int l1 = (t < L_) ? (t + 1) : L_;
      for (int l = l0; l < l1; ++l) {
        float2 f = fr[t - l];
        float2 c = cofsh[l];
        ar += c.x * f.x - c.y * f.y;
        ai += c.x * f.y + c.y * f.x;
      }
      ysum += ar * ar + ai * ai;
    }
    yr[it] = ar; yi[it] = ai;
  }
  red[tid] = ysum;
  __syncthreads();
  for (int off = 128; off; off >>= 1) {
    if (tid < off) red[tid] += red[tid + off];
    __syncthreads();
  }
  float ypwr = red[0] / (float)YLEN_;
  int snr = snrP[0];
  float npwr = ypwr * __powf(10.f, -0.1f * (float)snr);
  float sigma = sqrtf(npwr * 0.5f);
  if (tid == 0) out[NPW_OFF + np] = npwr;

  const float2* nz = noise + (size_t)np * YLEN_;
  float2* fo = frame + (size_t)np * SMK_;
  const float cfoK = 0.5f * 3.1415926f / 180.f;       // ANG*2*PI/360
#pragma unroll
  for (int it = 0; it < 19; ++it) {
    int t = tid + it * 256;
    if (t < SMK_) {
      float2 nu = nz[t];
      float ar = yr[it] + sigma * nu.x;
      float ai = yi[it] + sigma * nu.y;
      int pos = t % MK_;
      float ang = cfoK * (float)(pos - K_);
      float s, c;
      __sincosf(ang, &s, &c);
      fo[t] = make_float2(ar * c - ai * s, ar * s + ai * c);
    }
  }
}

// --------------------------------------------------------------------------
// K3: FFT-256 via WMMA DFT on CP-stripped symbols; write ip/isg to d_out.
// One block = 16 symbols out of N*P*17. Async-transpose staging.
// --------------------------------------------------------------------------
__global__ void k_fft_out(const float2* __restrict__ frame,
                          const float4* __restrict__ tw,
                          float* __restrict__ out) {
  __shared__ float2 Xlds[256 * 17];
  const int tid = threadIdx.x;
  const int g0 = blockIdx.x * 16;

  for (int sym = 0; sym < 16; ++sym) {
    int g = g0 + sym;
    int np = g / 17, s = g - np * 17;
    async_copy_f2(&frame[(size_t)np * SMK_ + s * MK_ + K_ + tid],
                  &Xlds[tid * 17 + sym]);
  }
  async_wait0();
  __syncthreads();

  const int wv = tid >> 5, lane = tid & 31;
  const int symL = lane & 15;
  const int kHalf = (lane >> 4) << 1;
  const int rowAdd = (lane >= 16) ? 8 : 0;
  const int g = g0 + symL;
  const int np = g / 17, s = g - np * 17;
  float* outBase = (s == 0)
      ? out + IP_OFF + (size_t)np * 256 * 2
      : out + IS_OFF + ((size_t)(np * 16 + (s - 1))) * 256 * 2;

  for (int tt = 0; tt < 2; ++tt) {
    const int mt = wv + tt * 8;
    v8f aCR = {0,0,0,0,0,0,0,0};
    v8f aSI = {0,0,0,0,0,0,0,0};
    v8f aCI = {0,0,0,0,0,0,0,0};
    v8f aSR = {0,0,0,0,0,0,0,0};
    const float4* twp = tw + (size_t)(mt * 64) * 32 + lane;
    for (int ks = 0; ks < 64; ++ks) {
      float4 w = twp[ks * 32];
      int kb = ks * 4 + kHalf;
      float2 b0 = Xlds[kb * 17 + symL];
      float2 b1 = Xlds[(kb + 1) * 17 + symL];
      v2f ac = {w.x, w.y}, as_ = {w.z, w.w};
      v2f br = {b0.x, b1.x}, bi = {b0.y, b1.y};
      aCR = wmma4(ac, br, aCR);
      aSI = wmma4(as_, bi, aSI);
      aCI = wmma4(ac, bi, aCI);
      aSR = wmma4(as_, br, aSR);
    }
    // fft: (c - i s)  ->  Re = cXr + sXi, Im = cXi - sXr
#pragma unroll
    for (int r = 0; r < 8; ++r) {
      int m = mt * 16 + rowAdd + r;
      outBase[m * 2]     = aCR[r] + aSI[r];
      outBase[m * 2 + 1] = aCI[r] - aSR[r];
    }
  }
}

// --------------------------------------------------------------------------
// K4: H_true[k] = sum_l cof[l] * exp(-2*pi*i*k*l/256)
// --------------------------------------------------------------------------
__global__ void k_htrue(const float2* __restrict__ cof,
                        float* __restrict__ out) {
  const int np = blockIdx.x, k = threadIdx.x;
  const float2* c = cof + np * L_;
  float hr = 0.f, hi = 0.f;
#pragma unroll
  for (int l = 0; l < L_; ++l) {
    float th = (float)((k * l) & 255) * TWO_PI_256;
    float s, cs;
    __sincosf(th, &s, &cs);
    float2 cv = c[l];
    hr += cv.x * cs + cv.y * s;
    hi += cv.y * cs - cv.x * s;
  }
  out[H_OFF + ((size_t)np * 256 + k) * 2]     = hr;
  out[H_OFF + ((size_t)np * 256 + k) * 2 + 1] = hi;
}

// --------------------------------------------------------------------------
extern "C" void kernel_launch(void* const* d_in, const int* in_sizes, int n_in,
                              void* d_out, int out_size, void* d_ws, size_t ws_size,
                              hipStream_t stream) {
  (void)in_sizes; (void)n_in; (void)out_size; (void)ws_size;
  const float2* x     = (const float2*)d_in[0];
  const float2* cof   = (const float2*)d_in[1];
  const float2* noise = (const float2*)d_in[2];
  const int*    snr   = (const int*)d_in[3];
  float* out = (float*)d_out;
  float* ws  = (float*)d_ws;

  float4* tw    = (float4*)ws;
  float2* pilot = (float2*)(ws + PILOT_OFF_F);
  float2* frame = (float2*)(ws + FRAME_OFF_F);

  k_twiddle   <<<128,  256, 0, stream>>>(tw);
  k_pilot     <<<1,    256, 0, stream>>>(pilot);
  k_ifft_frame<<<NP_,  256, 0, stream>>>(x, tw, frame);
  k_channel   <<<NP_,  256, 0, stream>>>(cof, noise, snr, pilot, frame, out);
  k_fft_out   <<<1088, 256, 0, stream>>>(frame, tw, out);
  k_htrue     <<<NP_,  256, 0, stream>>>(cof, out);
}